// GCN_61306363183616
// MI455X (gfx1250) — compile-verified
//
#include <hip/hip_runtime.h>
#include <hip/hip_bf16.h>

typedef __attribute__((ext_vector_type(2))) float v2f;
typedef __attribute__((ext_vector_type(4))) float v4f;
typedef __attribute__((ext_vector_type(8))) float v8f;

#define NNODES 50000
#define NEDGES 800000
#define IN_C   256
#define HID_C  128
#define OUT_C  64

// ---------------- degree / symmetric norm ----------------
__global__ __launch_bounds__(256)
void deg_init_kernel(float* __restrict__ deg, int n) {
  int i = blockIdx.x * blockDim.x + threadIdx.x;
  if (i < n) deg[i] = 1.0f;  // self loop contributes 1
}

__global__ __launch_bounds__(256)
void deg_edge_kernel(const long long* __restrict__ row, float* __restrict__ deg, int e) {
  int i = blockIdx.x * blockDim.x + threadIdx.x;
  if (i < e) atomicAdd(&deg[(int)row[i]], 1.0f);
}

__global__ __launch_bounds__(256)
void dis_kernel(float* __restrict__ deg, int n) {
  int i = blockIdx.x * blockDim.x + threadIdx.x;
  if (i < n) deg[i] = rsqrtf(deg[i]);   // deg >= 1 always
}

// ---------------- WMMA f32 GEMM: out[M,NC] = A[M,K] @ W[NC,K]^T + bias ----------------
// One wave computes one 16x16 output tile using V_WMMA_F32_16X16X4_F32.
// Wave32 fragment layout (ISA 7.12.2, 32-bit A 16x4 / B 4x16 / C 16x16):
//   A: lane 0-15 -> M=lane, K pair (k,k+1); lane 16-31 -> M=lane-16, K pair (k+2,k+3)
//   B: same striping with lane -> N column
//   D: VGPR i: lanes 0-15 M=i, lanes 16-31 M=i+8; N = lane&15
template<int K, int NC, bool RELU>
__global__ __launch_bounds__(256)
void gemm_wmma_kernel(const float* __restrict__ A, const float* __restrict__ W,
                      const float* __restrict__ bias, float* __restrict__ out,
                      int rowTiles) {
  const int lane    = threadIdx.x & 31;
  const int wave    = threadIdx.x >> 5;
  const int rowTile = blockIdx.x * 8 + wave;
  if (rowTile >= rowTiles) return;
  const int colTile = blockIdx.y;

  const int half = lane >> 4;   // selects K-pair within the fragment
  const int m16  = lane & 15;   // row (A) / column (B) index within tile

  const float* aptr = A + (size_t)(rowTile * 16 + m16) * K + half * 2;
  const float* bptr = W + (size_t)(colTile * 16 + m16) * K + half * 2; // W[n][k] row-major == B^T

  v8f acc = {};
#pragma unroll 8
  for (int k = 0; k < K; k += 4) {
    v2f a = *(const v2f*)(aptr + k);
    v2f b = *(const v2f*)(bptr + k);
    if (RELU) { a.x = fmaxf(a.x, 0.0f); a.y = fmaxf(a.y, 0.0f); }
    acc = __builtin_amdgcn_wmma_f32_16x16x4_f32(
        /*neg_a=*/false, a, /*neg_b=*/false, b,
        /*c_mod=*/(short)0, acc, /*reuse_a=*/false, /*reuse_b=*/false);
  }

  const float bv = bias[colTile * 16 + m16];
  float* optr = out + (size_t)(rowTile * 16 + half * 8) * NC + colTile * 16 + m16;
#pragma unroll
  for (int i = 0; i < 8; ++i)
    optr[(size_t)i * NC] = acc[i] + bv;
}

// ---------------- aggregation: out = selfloop term, then scatter-add edges ----------------
template<int C>
__global__ __launch_bounds__(256)
void selfloop_init_kernel(const float* __restrict__ h, const float* __restrict__ dis,
                          float* __restrict__ out, int n) {
  int idx = blockIdx.x * blockDim.x + threadIdx.x;
  if (idx >= n * C) return;
  int i = idx / C;
  float d = dis[i];
  out[idx] = d * d * h[idx];  // norm of self loop = dis[i]*dis[i]
}

template<int C>
__global__ __launch_bounds__(256)
void edge_agg_kernel(const long long* __restrict__ row, const long long* __restrict__ col,
                     const float* __restrict__ dis, const float* __restrict__ h,
                     float* __restrict__ out, int e) {
  const int lane = threadIdx.x & 31;
  const int eid  = (blockIdx.x << 3) + (threadIdx.x >> 5); // one wave per edge
  if (eid >= e) return;
  const int r = (int)row[eid];
  const int c = (int)col[eid];
  const float nrm = dis[r] * dis[c];
  constexpr int V = C / 32;  // channels per lane: 4 (C=128) or 2 (C=64)
  const float* src = h   + (size_t)r * C + lane * V;
  float*       dst = out + (size_t)c * C + lane * V;
  if constexpr (V == 4) {
    v4f v = *(const v4f*)src;
    atomicAdd(dst + 0, nrm * v.x);
    atomicAdd(dst + 1, nrm * v.y);
    atomicAdd(dst + 2, nrm * v.z);
    atomicAdd(dst + 3, nrm * v.w);
  } else {
    v2f v = *(const v2f*)src;
    atomicAdd(dst + 0, nrm * v.x);
    atomicAdd(dst + 1, nrm * v.y);
  }
}

// ---------------- log_softmax over 64 channels, one wave per node ----------------
__global__ __launch_bounds__(256)
void log_softmax_kernel(float* __restrict__ out, int n) {
  const int lane = threadIdx.x & 31;
  const int node = (blockIdx.x << 3) + (threadIdx.x >> 5);
  if (node >= n) return;
  float* p = out + (size_t)node * OUT_C + lane * 2;
  v2f v = *(const v2f*)p;
  float m = fmaxf(v.x, v.y);
#pragma unroll
  for (int off = 16; off > 0; off >>= 1)
    m = fmaxf(m, __shfl_xor(m, off, 32));
  float s = expf(v.x - m) + expf(v.y - m);
#pragma unroll
  for (int off = 16; off > 0; off >>= 1)
    s += __shfl_xor(s, off, 32);
  const float lse = m + logf(s);
  v.x -= lse; v.y -= lse;
  *(v2f*)p = v;
}

extern "C" void kernel_launch(void* const* d_in, const int* in_sizes, int n_in,
                              void* d_out, int out_size, void* d_ws, size_t ws_size,
                              hipStream_t stream) {
  (void)in_sizes; (void)n_in; (void)out_size; (void)ws_size;

  const float*     x    = (const float*)d_in[0];
  const long long* eidx = (const long long*)d_in[1];   // int64 [2, E]
  const float*     W1   = (const float*)d_in[2];       // [128, 256]
  const float*     b1   = (const float*)d_in[3];
  const float*     W2   = (const float*)d_in[4];       // [64, 128]
  const float*     b2   = (const float*)d_in[5];
  float*           out  = (float*)d_out;               // [N, 64]

  const long long* row = eidx;
  const long long* col = eidx + NEDGES;

  // workspace layout (floats): dis | h1 (reused as h2) | a1
  float* ws  = (float*)d_ws;
  float* dis = ws;                                  // N floats (deg -> rsqrt in place)
  float* h1  = ws + 50176;                          // N*128 (256B-aligned offset)
  float* a1  = h1 + (size_t)NNODES * HID_C;         // N*128
  float* h2  = h1;                                  // reuse: h1 dead after layer-1 agg

  // degree + symmetric normalization
  deg_init_kernel<<<(NNODES + 255) / 256, 256, 0, stream>>>(dis, NNODES);
  deg_edge_kernel<<<(NEDGES + 255) / 256, 256, 0, stream>>>(row, dis, NEDGES);
  dis_kernel<<<(NNODES + 255) / 256, 256, 0, stream>>>(dis, NNODES);

  const int rowTiles = NNODES / 16;  // 3125, exact

  // layer 1: h1 = x @ W1^T + b1   (WMMA f32)
  dim3 g1((rowTiles + 7) / 8, HID_C / 16);
  gemm_wmma_kernel<IN_C, HID_C, false><<<g1, 256, 0, stream>>>(x, W1, b1, h1, rowTiles);
  // a1 = selfloop + scatter-add over edges
  selfloop_init_kernel<HID_C><<<(NNODES * HID_C + 255) / 256, 256, 0, stream>>>(h1, dis, a1, NNODES);
  edge_agg_kernel<HID_C><<<(NEDGES + 7) / 8, 256, 0, stream>>>(row, col, dis, h1, a1, NEDGES);

  // layer 2: h2 = relu(a1) @ W2^T + b2   (ReLU fused into A loads)
  dim3 g2((rowTiles + 7) / 8, OUT_C / 16);
  gemm_wmma_kernel<HID_C, OUT_C, true><<<g2, 256, 0, stream>>>(a1, W2, b2, h2, rowTiles);
  selfloop_init_kernel<OUT_C><<<(NNODES * OUT_C + 255) / 256, 256, 0, stream>>>(h2, dis, out, NNODES);
  edge_agg_kernel<OUT_C><<<(NEDGES + 7) / 8, 256, 0, stream>>>(row, col, dis, h2, out, NNODES ? NEDGES : 0);

  // log_softmax in place on d_out
  log_softmax_kernel<<<(NNODES + 7) / 8, 256, 0, stream>>>(out, NNODES);
}